// RecSysGNN_36447092474029
// MI455X (gfx1250) — compile-verified
//
#include <hip/hip_runtime.h>
#include <hip/hip_bf16.h>

#define N_NODES 100000
#define N_EDGES 3200000
#define EMB_DIM 64
#define EPS_F 1e-16f

typedef __attribute__((ext_vector_type(2))) float v2f;
typedef __attribute__((ext_vector_type(8))) float v8f;

// ---- order-preserving float<->int key for integer atomicMax ----------------
__device__ __forceinline__ int f2key(float f) {
    int b = __float_as_int(f);
    return (b >= 0) ? b : (b ^ 0x7fffffff);
}
__device__ __forceinline__ float key2f(int k) {
    int b = (k >= 0) ? k : (k ^ 0x7fffffff);
    return __int_as_float(b);
}

// ---- init per-node segment state -------------------------------------------
__global__ void init_nodes(int* __restrict__ m_in, int* __restrict__ m_out,
                           float* __restrict__ s_in, float* __restrict__ s_out) {
    int i = blockIdx.x * blockDim.x + threadIdx.x;
    if (i < N_NODES) {
        m_in[i]  = (int)0x80000000;   // INT_MIN (== key below any real float)
        m_out[i] = (int)0x80000000;
        s_in[i]  = 0.0f;
        s_out[i] = 0.0f;
    }
}

// ---- segment max of logits over to_ (incoming) and from_ (outgoing) --------
__global__ void seg_max(const int* __restrict__ from_, const int* __restrict__ to_,
                        const float* __restrict__ attrs,
                        int* __restrict__ m_in, int* __restrict__ m_out) {
    int e = blockIdx.x * blockDim.x + threadIdx.x;
    if (e < N_EDGES) {
        int key = f2key(attrs[e]);          // S_TEMP == 1.0
        atomicMax(&m_in[to_[e]],  key);
        atomicMax(&m_out[from_[e]], key);
    }
}

// ---- segment sum of exp(l - m) ---------------------------------------------
__global__ void seg_sum(const int* __restrict__ from_, const int* __restrict__ to_,
                        const float* __restrict__ attrs,
                        const int* __restrict__ m_in, const int* __restrict__ m_out,
                        float* __restrict__ s_in, float* __restrict__ s_out) {
    int e = blockIdx.x * blockDim.x + threadIdx.x;
    if (e < N_EDGES) {
        float l = attrs[e];
        int t = to_[e], f = from_[e];
        atomicAdd(&s_in[t],  expf(l - key2f(m_in[t])));
        atomicAdd(&s_out[f], expf(l - key2f(m_out[f])));
    }
}

// ---- per-edge norm = sqrt(softmax_in * softmax_out) (closed form) ----------
__global__ void edge_norm(const int* __restrict__ from_, const int* __restrict__ to_,
                          const float* __restrict__ attrs,
                          const int* __restrict__ m_in, const int* __restrict__ m_out,
                          const float* __restrict__ s_in, const float* __restrict__ s_out,
                          float* __restrict__ norm) {
    int e = blockIdx.x * blockDim.x + threadIdx.x;
    if (e < N_EDGES) {
        float l = attrs[e];
        int t = to_[e], f = from_[e];
        float mi = key2f(m_in[t]);
        float mo = key2f(m_out[f]);
        float denom = (s_in[t] + EPS_F) * (s_out[f] + EPS_F);
        norm[e] = expf(l - 0.5f * (mi + mo)) / sqrtf(denom);
    }
}

// ---- utility: zero / copy in float4 ----------------------------------------
__global__ void zero4(float4* __restrict__ p, int n4) {
    int i = blockIdx.x * blockDim.x + threadIdx.x;
    if (i < n4) p[i] = make_float4(0.f, 0.f, 0.f, 0.f);
}
__global__ void copy4(const float4* __restrict__ s, float4* __restrict__ d, int n4) {
    int i = blockIdx.x * blockDim.x + threadIdx.x;
    if (i < n4) d[i] = s[i];
}

// ---- one propagation layer: next[to] += norm * cur[from] -------------------
// One wave per edge (grid-stride). Lane d handles dims {2d, 2d+1} as float2:
// a single fully-coalesced 256B row gather per edge, then 64 f32 atomics.
__global__ void scatter_layer(const int* __restrict__ from_, const int* __restrict__ to_,
                              const float* __restrict__ norm,
                              const float* __restrict__ cur, float* __restrict__ nxt) {
    int lane   = threadIdx.x & 31;
    int warp   = blockIdx.x * (blockDim.x >> 5) + (threadIdx.x >> 5);
    int nwarps = gridDim.x * (blockDim.x >> 5);
    const float2* cur2 = (const float2*)cur;
    for (int e = warp; e < N_EDGES; e += nwarps) {
        int f = from_[e];
        int t = to_[e];
        float w = norm[e];
        float2 v = cur2[(size_t)f * 32 + lane];   // 256B coalesced row gather
        float* dst = nxt + (size_t)t * EMB_DIM + lane * 2;
        atomicAdd(dst,     w * v.x);
        atomicAdd(dst + 1, w * v.y);
    }
}

// ---- final layer mean as a dense GEMM on the matrix pipe -------------------
// out = 0.25*(m0 + m1 + m2 + m3) computed as [m0|m1|m2|m3](N x 256) * W(256 x 64),
// W = 4 stacked 0.25*I64 blocks, via v_wmma_f32_16x16x4_f32 accumulation.
// One wave per (16-node tile, 16-dim tile): 4 sources * 4 K-chunks = 16 WMMAs.
// A layout (ISA 16x4 f32): lane m=lane&15; VGPR0 holds K=2*half, VGPR1 K=2*half+1.
// B (4x16, mirrored layout) is the 0.25-identity chunk built in-register.
// D layout (16x16 f32): VGPR v -> row v + 8*half, col = lane&15.
__global__ void wmma_mean(const float* __restrict__ m0, const float* __restrict__ m1,
                          const float* __restrict__ m2, const float* __restrict__ m3,
                          float* __restrict__ out) {
    int lane = threadIdx.x & 31;
    int warp = blockIdx.x * (blockDim.x >> 5) + (threadIdx.x >> 5);
    int nodeTile = warp >> 2;          // 0 .. 6249  (100000/16 tiles)
    int dimTile  = warp & 3;           // 0 .. 3     (4 x 16 dims)
    int half = lane >> 4;
    int n    = lane & 15;
    int row  = nodeTile * 16 + n;      // A-matrix row index (M = lane&15)
    int col0 = dimTile * 16;

    const float* srcs[4] = {m0, m1, m2, m3};
    v8f c = {};
    for (int s = 0; s < 4; ++s) {
        const float* rp = srcs[s] + (size_t)row * EMB_DIM + col0;
#pragma unroll
        for (int j = 0; j < 4; ++j) {
            int k0 = 4 * j;
            v2f a, b;
            a.x = rp[k0 + 2 * half];                        // K = 2*half
            a.y = rp[k0 + 2 * half + 1];                    // K = 2*half+1
            b.x = (n == k0 + 2 * half)     ? 0.25f : 0.0f;  // 0.25*I chunk
            b.y = (n == k0 + 2 * half + 1) ? 0.25f : 0.0f;
            c = __builtin_amdgcn_wmma_f32_16x16x4_f32(
                    /*neg_a=*/false, a, /*neg_b=*/false, b,
                    /*c_mod=*/(short)0, c, /*reuse_a=*/false, /*reuse_b=*/false);
        }
    }
    // store D: VGPR v -> node row (nodeTile*16 + v + 8*half), column col0 + n
    float* dst = out + (size_t)(nodeTile * 16 + 8 * half) * EMB_DIM + col0 + n;
#pragma unroll
    for (int v = 0; v < 8; ++v) dst[(size_t)v * EMB_DIM] = c[v];
}

extern "C" void kernel_launch(void* const* d_in, const int* in_sizes, int n_in,
                              void* d_out, int out_size, void* d_ws, size_t ws_size,
                              hipStream_t stream) {
    const float* emb   = (const float*)d_in[0];        // [N_NODES, 64]
    const int*   ei    = (const int*)d_in[1];          // [2, N_EDGES]
    const int*   from_ = ei;
    const int*   to_   = ei + N_EDGES;
    const float* attrs = (const float*)d_in[2];        // [N_EDGES]
    float* out = (float*)d_out;                        // emb copy | mean result

    // carve workspace (~91 MB)
    float* norm  = (float*)d_ws;                       // E
    int*   m_in  = (int*)(norm + N_EDGES);             // N
    int*   m_out = m_in + N_NODES;                     // N
    float* s_in  = (float*)(m_out + N_NODES);          // N
    float* s_out = s_in + N_NODES;                     // N
    float* c1    = s_out + N_NODES;                    // N*64
    float* c2    = c1 + (size_t)N_NODES * EMB_DIM;     // N*64
    float* c3    = c2 + (size_t)N_NODES * EMB_DIM;     // N*64

    const int TB = 256;
    const int nodeBlocks = (N_NODES + TB - 1) / TB;
    const int edgeBlocks = (N_EDGES + TB - 1) / TB;
    const int n4 = N_NODES * EMB_DIM / 4;
    const int v4Blocks = (n4 + TB - 1) / TB;

    // 1) segment softmax pieces (closed-form norm)
    init_nodes<<<nodeBlocks, TB, 0, stream>>>(m_in, m_out, s_in, s_out);
    seg_max  <<<edgeBlocks, TB, 0, stream>>>(from_, to_, attrs, m_in, m_out);
    seg_sum  <<<edgeBlocks, TB, 0, stream>>>(from_, to_, attrs, m_in, m_out, s_in, s_out);
    edge_norm<<<edgeBlocks, TB, 0, stream>>>(from_, to_, attrs, m_in, m_out, s_in, s_out, norm);

    // 2) three propagation layers: emb -> c1 -> c2 -> c3
    const float* cur = emb;
    float* bufs[3] = {c1, c2, c3};
    for (int layer = 0; layer < 3; ++layer) {
        float* nxt = bufs[layer];
        zero4<<<v4Blocks, TB, 0, stream>>>((float4*)nxt, n4);
        scatter_layer<<<12800, TB, 0, stream>>>(from_, to_, norm, cur, nxt);
        cur = nxt;
    }

    // 3) outputs: (emb, mean of [emb, c1, c2, c3])
    copy4<<<v4Blocks, TB, 0, stream>>>((const float4*)emb, (float4*)out, n4);
    // 25000 warps (6250 node tiles x 4 dim tiles), 8 waves/block -> 3125 blocks
    wmma_mean<<<3125, 256, 0, stream>>>(emb, c1, c2, c3, out + (size_t)N_NODES * EMB_DIM);
}